// FS_Layer_15633680957473
// MI455X (gfx1250) — compile-verified
//
#include <hip/hip_runtime.h>

// FS layer: out[b, r] = prod_i (mv[b,i,m_i] + eps), r in product order (input 0 slowest).
// Factor r = hi*256 + lo: hi over inputs 0..2 (64 combos), lo over inputs 3..6 (256 combos).
// out row (64 x 256) = P_hi (64x1) x P_lo (1x256) -> rank-1 outer product via
// V_WMMA_F32_16X16X4_F32 tiles: K=0 slots carry the vectors; A's K=2 lanes are masked
// to 0.0 so B's K=2 lanes may hold finite garbage (0 * finite = 0 in the FMA chain).
// Store-bandwidth bound: 268 MB out @ 23.3 TB/s ~= 11.5 us floor; NT stores bypass
// the 192 MB L2 since the output streams straight through.

typedef __attribute__((ext_vector_type(2))) float v2f;
typedef __attribute__((ext_vector_type(8))) float v8f;

#define FS_EPS 1e-6f
#define LO_DIM 256   // 4^4, inputs 3..6
#define HI_DIM 64    // 4^3, inputs 0..2
#define R_TOTAL 16384

__global__ __launch_bounds__(256) void fs_outer_wmma_kernel(
    const float* __restrict__ mv,   // [B, 7, 4]
    float* __restrict__ out,        // [B, 16384]
    int B) {
  __shared__ float s_mv[28];
  __shared__ float s_hi[HI_DIM];
  __shared__ float s_lo[LO_DIM];

  const int t = threadIdx.x;
  const int row = blockIdx.x;
  if (row >= B) return;

  // Load the 28 membership values (+eps) for this row into LDS.
  if (t < 28) s_mv[t] = mv[(size_t)row * 28 + t] + FS_EPS;
  __syncthreads();

  // P_lo[t]: product over inputs 3..6, m3 varies slowest within lo.
  {
    float p = s_mv[12 + ((t >> 6) & 3)];
    p *= s_mv[16 + ((t >> 4) & 3)];
    p *= s_mv[20 + ((t >> 2) & 3)];
    p *= s_mv[24 + (t & 3)];
    s_lo[t] = p;
  }
  // P_hi[t]: product over inputs 0..2, m0 varies slowest overall.
  if (t < HI_DIM) {
    float p = s_mv[0 + ((t >> 4) & 3)];
    p *= s_mv[4 + ((t >> 2) & 3)];
    p *= s_mv[8 + (t & 3)];
    s_hi[t] = p;
  }
  __syncthreads();

  const int wave = t >> 5;       // 8 waves per block
  const int lane = t & 31;
  const int half = lane >> 4;    // 0: lanes 0-15 (K=0 slots), 1: lanes 16-31 (K=2 slots)
  const int l16  = lane & 15;

  // Per-wave tile strip: tiles wave*8 .. wave*8+7  ->  H constant, L = Lbase..Lbase+7.
  const int H     = wave >> 1;        // 0..3  (hi tile, fixed per wave)
  const int Lbase = (wave & 1) * 8;   // 0 or 8 (lo tile start)

  // A fragment is loop-invariant: VGPR0 lanes 0-15 = (M=lane, K=0) = P_hi; K=2 lanes
  // masked to exact 0.0 via constant select (no predicated LDS load, EXEC stays full).
  const float amask = (half == 0) ? 1.0f : 0.0f;
  v2f a;
  a.x = s_hi[H * 16 + l16] * amask;   // unconditional ds_load, then v_mul mask
  a.y = 0.0f;                         // K=1 / K=3 slots

  // One base pointer; all 64 store offsets become compile-time immediates.
  float* base = out + (size_t)row * R_TOTAL
                    + (size_t)(H * 16 + half * 8) * LO_DIM + Lbase * 16 + l16;
  const float* slo = &s_lo[Lbase * 16 + l16];   // ds base; per-tile offset = k*64 bytes

  #pragma unroll
  for (int k = 0; k < 8; ++k) {
    // B fragment: VGPR0 lanes 0-15 = (K=0, N=lane) = P_lo; lanes 16-31 carry finite
    // duplicates in the K=2 slot, annihilated by A's zeroed K=2 column.
    v2f b;
    b.x = slo[k * 16];                // ds_load_b32 with immediate offset
    b.y = 0.0f;

    v8f c = {};
    // D[m][n] = P_hi[H*16+m] * P_lo[(Lbase+k)*16+n]  (rank-1; FMA adds only exact zeros)
    c = __builtin_amdgcn_wmma_f32_16x16x4_f32(
        /*neg_a=*/false, a, /*neg_b=*/false, b,
        /*c_mod=*/(short)0, c, /*reuse_a=*/false, /*reuse_b=*/false);

    // D layout: VGPR v, lanes 0-15 -> M=v, N=lane; lanes 16-31 -> M=v+8, N=lane-16.
    // Memory: base + k*16 (lo tile) + v*256 (hi row); contiguous 64B lane segments.
    #pragma unroll
    for (int v = 0; v < 8; ++v) {
      __builtin_nontemporal_store(c[v], base + k * 16 + (size_t)v * LO_DIM);
    }
  }
}

extern "C" void kernel_launch(void* const* d_in, const int* in_sizes, int n_in,
                              void* d_out, int out_size, void* d_ws, size_t ws_size,
                              hipStream_t stream) {
  const float* mv = (const float*)d_in[0];   // [B, 7, 4] f32
  float* out = (float*)d_out;                // [B, 16384] f32
  const int B = in_sizes[0] / 28;            // 4096

  fs_outer_wmma_kernel<<<dim3(B), dim3(256), 0, stream>>>(mv, out, B);
}